// HBVMulTDET_WaterLoss_53781580480885
// MI455X (gfx1250) — compile-verified
//
#include <hip/hip_runtime.h>

#define NEARZERO 1e-5f
#define NMUL  16
#define NSTEP 365
#define NGRID 10000
#define BLOCK 256
#define GPB   (BLOCK / NMUL)   /* 16 grid cells per block */
#define ROWF  (GPB * 3)        /* 48 floats of forcing per block per step */
#define CHUNK 8
#define NCHUNKS ((NSTEP + CHUNK - 1) / CHUNK)

#if defined(__gfx1250__) && __has_builtin(__builtin_amdgcn_global_load_async_to_lds_b128)
#define HAVE_ASYNC 1
#else
#define HAVE_ASYNC 0
#endif

#if defined(__gfx1250__) && __has_builtin(__builtin_amdgcn_s_wait_asynccnt)
#define WAIT_ASYNC(n) __builtin_amdgcn_s_wait_asynccnt(n)
#elif defined(__gfx1250__)
#define WAIT_ASYNC(n) asm volatile("s_wait_asynccnt %0" ::"i"(n) : "memory")
#else
#define WAIT_ASYNC(n)
#endif

// Fast transcendentals straight on the hardware ops (v_log_f32 / v_exp_f32).
__device__ __forceinline__ float fast_log2(float x) { return __builtin_amdgcn_logf(x); }
__device__ __forceinline__ float fast_exp2(float x) { return __builtin_amdgcn_exp2f(x); }
__device__ __forceinline__ float powf_pos(float x, float y) {
  // x > 0 guaranteed (SM >= NEARZERO, FC >= 50): pow = exp2(y*log2(x))
  return fast_exp2(y * fast_log2(x));
}
__device__ __forceinline__ float fast_exp(float x) {
  return fast_exp2(x * 1.4426950408889634f);
}

#if HAVE_ASYNC
typedef int v4i __attribute__((vector_size(16)));
typedef __attribute__((address_space(1))) v4i* gbl_v4i;  // pointer to AS1 v4i32
typedef __attribute__((address_space(3))) v4i* lds_v4i;  // pointer to AS3 v4i32
#endif

__global__ __launch_bounds__(BLOCK)
void hbv_wl_kernel(const float* __restrict__ x,     // (365,10000,3) P,T,PET
                   const float* __restrict__ praw,  // (365,10000,3,16)
                   const float* __restrict__ wraw,  // (10000,13,16)
                   const float* __restrict__ ac,    // (10000,)
                   float* __restrict__ out) {       // (365,10000)
  const int tid = threadIdx.x;
  const int gid = blockIdx.x * BLOCK + tid;   // grid is exact: 625*256 = 160000
  const int g = gid >> 4;
  const int m = gid & (NMUL - 1);

  // ---- time-invariant parameter scaling (hoisted out of the recurrence) ----
  const size_t pbase = (size_t)(NSTEP - 1) * NGRID * 3 * NMUL + (size_t)g * 3 * NMUL + m;
  const float parBETA   = 1.0f  + praw[pbase + 0 * NMUL] * 5.0f;
  const float parK0     = 0.05f + praw[pbase + 1 * NMUL] * 0.85f;
  const float parBETAET = 0.3f  + praw[pbase + 2 * NMUL] * 4.7f;

  const size_t wbase = (size_t)g * 13 * NMUL + m;
  const float parFC    = 50.0f  + wraw[wbase +  0 * NMUL] * 950.0f;
  const float parK1    = 0.01f  + wraw[wbase +  1 * NMUL] * 0.49f;
  const float parK2    = 0.001f + wraw[wbase +  2 * NMUL] * 0.199f;
  const float parLP    = 0.2f   + wraw[wbase +  3 * NMUL] * 0.8f;
  const float parPERC  =          wraw[wbase +  4 * NMUL] * 10.0f;
  const float parUZL   =          wraw[wbase +  5 * NMUL] * 100.0f;
  const float parTT    = -2.5f  + wraw[wbase +  6 * NMUL] * 5.0f;
  const float parCFMAX = 0.5f   + wraw[wbase +  7 * NMUL] * 9.5f;
  const float parCFR   =          wraw[wbase +  8 * NMUL] * 0.1f;
  const float parCWH   =          wraw[wbase +  9 * NMUL] * 0.2f;
  const float parC     =          wraw[wbase + 10 * NMUL] * 1.0f;
  const float parTR    =          wraw[wbase + 11 * NMUL] * 20.0f;
  const float parAc    =          wraw[wbase + 12 * NMUL] * 2500.0f;

  const float acm = ac[g];
  const float rf  = fminf(fmaxf((acm - parAc) * 1e-3f, -1.0f), 1.0f);
  const float ex  = fast_exp(fminf(fmaxf((2500.0f - acm) * 0.02f, -10.0f), 0.0f));
  const float regional = (acm < 2500.0f) ? rf * parTR : ex * parTR;

  const float invFC    = 1.0f / parFC;
  const float invLPFC  = 1.0f / (parLP * parFC);
  const float cfrcfmax = parCFR * parCFMAX;

  float SNOWPACK = 0.001f, MELTWATER = 0.001f, SM = 0.001f, SUZ = 0.001f, SLZ = 0.001f;

  auto step = [&](float Pm, float Tm, float PETm, int t) {
    const bool  warm = (Tm >= parTT);
    const float RAIN = warm ? Pm : 0.0f;
    SNOWPACK += warm ? 0.0f : Pm;
    const float melt = fminf(fmaxf(parCFMAX * (Tm - parTT), 0.0f), SNOWPACK);
    MELTWATER += melt; SNOWPACK -= melt;
    const float refr = fminf(fmaxf(cfrcfmax * (parTT - Tm), 0.0f), MELTWATER);
    SNOWPACK += refr; MELTWATER -= refr;
    const float tosoil = fmaxf(MELTWATER - parCWH * SNOWPACK, 0.0f);
    MELTWATER -= tosoil;
    const float sw = fminf(powf_pos(SM * invFC, parBETA), 1.0f);
    const float rt = RAIN + tosoil;
    const float recharge = rt * sw;
    SM += rt - recharge;
    const float excess = fmaxf(SM - parFC, 0.0f);
    SM -= excess;
    const float ef = fminf(powf_pos(SM * invLPFC, parBETAET), 1.0f);
    const float ETact = fminf(SM, PETm * ef);
    SM = fmaxf(SM - ETact, NEARZERO);
    const float cap = fminf(SLZ, parC * SLZ * (1.0f - fminf(SM * invFC, 1.0f)));
    SM  = fmaxf(SM + cap, NEARZERO);
    SLZ = fmaxf(SLZ - cap, NEARZERO);
    SUZ += recharge + excess;
    const float perc = fminf(SUZ, parPERC); SUZ -= perc;
    const float Q0 = parK0 * fmaxf(SUZ - parUZL, 0.0f); SUZ -= Q0;
    const float Q1 = parK1 * SUZ; SUZ -= Q1;
    SLZ += perc;
    SLZ = fmaxf(SLZ + regional, 0.0f);
    const float Q2 = parK2 * SLZ; SLZ -= Q2;
    // wave32: the 16 muls of one grid cell are 16 contiguous lanes -> butterfly
    float q = Q0 + Q1 + Q2;
    q += __shfl_xor(q, 1, 32);
    q += __shfl_xor(q, 2, 32);
    q += __shfl_xor(q, 4, 32);
    q += __shfl_xor(q, 8, 32);
    if (m == 0) out[(size_t)t * NGRID + g] = q * (1.0f / 16.0f);
  };

#if HAVE_ASYNC
  // Double-buffered async DMA staging of forcings: each block's 16 grid cells
  // are 192 contiguous, 16B-aligned bytes per step (x base 256B aligned,
  // step stride 120000B = 16*7500, block offset 192B). One 8-step chunk =
  // 96 x b128 lane transfers (threads 0..95, waves 0-2), tracked by ASYNCcnt.
  __shared__ __align__(16) float sbuf[2][CHUNK * ROWF];
  const int    gl3    = (tid >> 4) * 3;
  const size_t blkoff = (size_t)blockIdx.x * ROWF;

  auto issue_chunk = [&](int t0, int b) {
    if (tid < CHUNK * (ROWF / 4)) {        // 96 lanes, one b128 each
      const int sic = tid / (ROWF / 4);    // step within chunk
      const int seg = tid % (ROWF / 4);    // 16-byte segment within step row
      const int t = t0 + sic;
      if (t < NSTEP) {
        const float* gsrc = x + (size_t)t * (NGRID * 3) + blkoff + seg * 4;
        const float* ldst = &sbuf[b][sic * ROWF + seg * 4];
        __builtin_amdgcn_global_load_async_to_lds_b128(
            (gbl_v4i)(void*)gsrc,
            (lds_v4i)(void*)ldst,
            0, 0);
      }
    }
  };

  issue_chunk(0, 0);
  for (int c = 0; c < NCHUNKS; ++c) {
    const int t0 = c * CHUNK;
    if (c + 1 < NCHUNKS) { issue_chunk(t0 + CHUNK, (c + 1) & 1); WAIT_ASYNC(1); }
    else                 { WAIT_ASYNC(0); }
    __syncthreads();                       // LDS chunk c visible to all waves
    const float* row = sbuf[c & 1];
    const int ns = min(CHUNK, NSTEP - t0);
    for (int s = 0; s < ns; ++s) {
      const float Pm = row[s * ROWF + gl3 + 0];  // 16-lane LDS broadcast
      const float Tm = row[s * ROWF + gl3 + 1];
      const float Em = row[s * ROWF + gl3 + 2];
      step(Pm, Tm, Em, t0 + s);
    }
    __syncthreads();                       // done reading before buffer reuse
  }
#else
  // Fallback: direct per-thread loads (wave-coalesced) + global_prefetch_b8
  const float* xp = x + (size_t)g * 3;
  for (int t = 0; t < NSTEP; ++t) {
    const float* p = xp + (size_t)t * (NGRID * 3);
    if (t + 8 < NSTEP) __builtin_prefetch(p + (size_t)8 * NGRID * 3, 0, 0);
    step(p[0], p[1], p[2], t);
  }
#endif
}

extern "C" void kernel_launch(void* const* d_in, const int* in_sizes, int n_in,
                              void* d_out, int out_size, void* d_ws, size_t ws_size,
                              hipStream_t stream) {
  const float* x    = (const float*)d_in[0];  // x_hydro_model (365,10000,3)
  const float* praw = (const float*)d_in[1];  // params_raw (365,10000,3,16)
  const float* wraw = (const float*)d_in[2];  // waterloss_params_raw (10000,13,16)
  const float* ac   = (const float*)d_in[3];  // ac_batch (10000,)
  float* out = (float*)d_out;                 // (365,10000)
  (void)in_sizes; (void)n_in; (void)out_size; (void)d_ws; (void)ws_size;

  const int nthreads = NGRID * NMUL;          // 160000 = 625 * 256 exactly
  hbv_wl_kernel<<<dim3(nthreads / BLOCK), dim3(BLOCK), 0, stream>>>(x, praw, wraw, ac, out);
}